// DSEA_54460185313445
// MI455X (gfx1250) — compile-verified
//
#include <hip/hip_runtime.h>
#include <math.h>

// ---------------- problem constants (match reference) ----------------
#define D_N   15000
#define D_E   200000
#define D_EA  400000
#define D_R   1000
#define D_B   512
#define D_EHD 300
#define D_RHD 100

typedef float v2f __attribute__((ext_vector_type(2)));
typedef float v8f __attribute__((ext_vector_type(8)));

__device__ __forceinline__ float dsea_leaky(float x){ return x > 0.f ? x : 0.01f * x; }
__device__ __forceinline__ float dsea_sig(float x){ return 1.f / (1.f + expf(-x)); }
__device__ __forceinline__ float dsea_act(float x, int a){
  if (a == 1) return fmaxf(x, 0.f);
  if (a == 2) return dsea_leaky(x);
  if (a == 3) return dsea_sig(x);
  return x;
}
__device__ __forceinline__ void dsea_atomic_max_f(float* addr, float v){
  int* ai = (int*)addr;
  int old = *ai;
  while (__int_as_float(old) < v) {
    int assumed = old;
    old = atomicCAS(ai, assumed, __float_as_int(v));
    if (old == assumed) break;
  }
}

// ---------------- generic fp32 WMMA GEMM ----------------
// C[M,Nc] = ACT( sum_k A[m,k]*B(k,n) + bias[n] + cadd + addsrc[m,n] )
// BL==0 : B stored (Nc x K) row-major ("x @ W^T"):  B(k,n) = Bm[n*ldb + k]
// BL==1 : B stored (K x Nc) row-major ("x @ W"):    B(k,n) = Bm[k*ldb + n]
// One wave32 computes a 16 x 64 strip (NT=4 accumulators) of C via
// V_WMMA_F32_16X16X4_F32, K stepped by 4.  Fragment layouts per ISA 7.12.2:
// A frag (16x4):  lane<16 -> M=lane, K={k0,k0+1}; lane>=16 -> M=lane-16, K={k0+2,k0+3}
// B frag (4x16):  lane<16 -> N=lane, K={k0,k0+1}; lane>=16 -> N=lane-16, K={k0+2,k0+3}
// C frag (16x16): vgpr v: lane<16 -> (M=v,   N=lane), lane>=16 -> (M=v+8, N=lane-16)
template<int BL, int ACT>
__global__ __launch_bounds__(256)
void dsea_gemm(const float* __restrict__ A, int lda,
               const float* __restrict__ Bm, int ldb,
               float* __restrict__ C, int ldc,
               const float* __restrict__ bias,
               const float* __restrict__ addsrc, int ldadd,
               float cadd, int M, int Nc, int K)
{
  constexpr int NT = 4;
  const v8f vzero = {0.f,0.f,0.f,0.f,0.f,0.f,0.f,0.f};
  int wave = threadIdx.x >> 5;
  int lane = threadIdx.x & 31;
  int tN = (Nc + 15) >> 4;
  int tM = (M + 15) >> 4;
  int tNg = (tN + NT - 1) / NT;
  int grp = blockIdx.x * (blockDim.x >> 5) + wave;
  if (grp >= tM * tNg) return;
  int tm = grp / tNg;
  int tn0 = (grp - tm * tNg) * NT;
  int lo = lane & 15, hi = lane >> 4;
  int Kmain = K & ~3;

  if (tm * 16 + 16 <= M && (tn0 + NT) * 16 <= Nc) {
    // ---------- fast path: fully interior 16 x 64 strip, no guards ----------
    const float* Arow = A + (size_t)(tm * 16 + lo) * lda + (hi << 1);
    const float* Brow[NT];
#pragma unroll
    for (int t = 0; t < NT; ++t) {
      if (BL == 0) Brow[t] = Bm + (size_t)((tn0 + t) * 16 + lo) * ldb + (hi << 1);
      else         Brow[t] = Bm + (size_t)(hi << 1) * ldb + (tn0 + t) * 16 + lo;
    }
    v8f acc[NT];
#pragma unroll
    for (int t = 0; t < NT; ++t) acc[t] = vzero;
    for (int k0 = 0; k0 < Kmain; k0 += 4) {
      if (((k0 & 63) == 0) && (k0 + 64 < K))
        __builtin_prefetch(Arow + k0 + 64, 0, 1);
      v2f av = *(const v2f*)(Arow + k0);
#pragma unroll
      for (int t = 0; t < NT; ++t) {
        v2f bv;
        if (BL == 0) {
          bv = *(const v2f*)(Brow[t] + k0);
        } else {
          bv[0] = Brow[t][(size_t)k0 * ldb];
          bv[1] = Brow[t][(size_t)(k0 + 1) * ldb];
        }
        acc[t] = __builtin_amdgcn_wmma_f32_16x16x4_f32(false, av, false, bv,
                                                       (short)0, acc[t], false, false);
      }
    }
    if (Kmain < K) {           // K tail (K%4 != 0): one guarded step
      int ka = Kmain + (hi << 1);
      int k1 = ka < K - 1 ? ka : K - 1;
      int k2 = ka + 1 < K - 1 ? ka + 1 : K - 1;
      v2f av;
      float a0 = Arow[k1 - (hi << 1)];
      float a1 = Arow[k2 - (hi << 1)];
      av[0] = (ka     < K) ? a0 : 0.f;
      av[1] = (ka + 1 < K) ? a1 : 0.f;
#pragma unroll
      for (int t = 0; t < NT; ++t) {
        float b0, b1;
        if (BL == 0) {
          const float* bb = Brow[t] - (hi << 1);
          b0 = bb[k1]; b1 = bb[k2];
        } else {
          const float* bb = Brow[t] - (size_t)(hi << 1) * ldb;
          b0 = bb[(size_t)k1 * ldb]; b1 = bb[(size_t)k2 * ldb];
        }
        v2f bv;
        bv[0] = (ka     < K) ? b0 : 0.f;
        bv[1] = (ka + 1 < K) ? b1 : 0.f;
        acc[t] = __builtin_amdgcn_wmma_f32_16x16x4_f32(false, av, false, bv,
                                                       (short)0, acc[t], false, false);
      }
    }
#pragma unroll
    for (int t = 0; t < NT; ++t) {
      int ccol = (tn0 + t) * 16 + lo;
      float bval = bias ? bias[ccol] : 0.f;
#pragma unroll
      for (int v = 0; v < 8; ++v) {
        int crow = tm * 16 + v + (hi << 3);
        float x = acc[t][v] + bval + cadd;
        if (addsrc) x += addsrc[(size_t)crow * ldadd + ccol];
        C[(size_t)crow * ldc + ccol] = dsea_act(x, ACT);
      }
    }
  } else {
    // ---------- edge path: clamped (in-bounds) loads + v_cndmask zeroing ----------
    for (int t = 0; t < NT; ++t) {
      int tn = tn0 + t;
      if (tn >= tN) break;
      int arow = tm * 16 + lo;
      int bcol = tn * 16 + lo;
      int arc = arow < M - 1 ? arow : M - 1;
      int bcc = bcol < Nc - 1 ? bcol : Nc - 1;
      bool aok = arow < M, bok = bcol < Nc;
      v8f acc = vzero;
      for (int k0 = 0; k0 < K; k0 += 4) {
        int ka = k0 + (hi << 1);
        int k1 = ka < K - 1 ? ka : K - 1;
        int k2 = ka + 1 < K - 1 ? ka + 1 : K - 1;
        float a0 = A[(size_t)arc * lda + k1];
        float a1 = A[(size_t)arc * lda + k2];
        float b0, b1;
        if (BL == 0) { b0 = Bm[(size_t)bcc * ldb + k1]; b1 = Bm[(size_t)bcc * ldb + k2]; }
        else         { b0 = Bm[(size_t)k1 * ldb + bcc]; b1 = Bm[(size_t)k2 * ldb + bcc]; }
        v2f av, bv;
        av[0] = (aok && ka     < K) ? a0 : 0.f;
        av[1] = (aok && ka + 1 < K) ? a1 : 0.f;
        bv[0] = (bok && ka     < K) ? b0 : 0.f;
        bv[1] = (bok && ka + 1 < K) ? b1 : 0.f;
        acc = __builtin_amdgcn_wmma_f32_16x16x4_f32(false, av, false, bv,
                                                    (short)0, acc, false, false);
      }
      if (bok) {
        int ccol = bcol;
        float bval = bias ? bias[ccol] : 0.f;
#pragma unroll
        for (int v = 0; v < 8; ++v) {
          int crow = tm * 16 + v + (hi << 3);
          if (crow < M) {
            float x = acc[v] + bval + cadd;
            if (addsrc) x += addsrc[(size_t)crow * ldadd + ccol];
            C[(size_t)crow * ldc + ccol] = dsea_act(x, ACT);
          }
        }
      }
    }
  }
}

// ---------------- simple utility kernels ----------------
__global__ void dsea_fill(float* p, float v, long long n){
  long long i = (long long)blockIdx.x * blockDim.x + threadIdx.x;
  if (i < n) p[i] = v;
}
__global__ void dsea_fill_i(int* p, int v, long long n){
  long long i = (long long)blockIdx.x * blockDim.x + threadIdx.x;
  if (i < n) p[i] = v;
}
__global__ void dsea_vecadd2(const float* a, const float* b, float* d, int n){
  int i = blockIdx.x * blockDim.x + threadIdx.x;
  if (i < n) d[i] = a[i] + b[i];
}
__global__ void dsea_count(const int* idx, float* deg, int L){
  int l = blockIdx.x * blockDim.x + threadIdx.x;
  if (l < L) atomicAdd(&deg[idx[l]], 1.f);
}
__global__ void dsea_dis(const float* deg, float* dis, int n){
  int i = blockIdx.x * blockDim.x + threadIdx.x;
  if (i < n) dis[i] = deg[i] > 0.f ? 1.f / sqrtf(deg[i]) : 0.f;
}
// GCN scatter: out[i,:] += dis[j]*dis[i]*x[j,:]   (j=src=eia[l], i=tgt=eia[L+l])
__global__ void dsea_gcn_scatter(const float* __restrict__ x, const int* __restrict__ eia,
                                 const float* __restrict__ dis, float* __restrict__ out, int L){
  long long tid = (long long)blockIdx.x * blockDim.x + threadIdx.x;
  if (tid >= (long long)L * D_EHD) return;
  int l = (int)(tid / D_EHD), d = (int)(tid - (long long)l * D_EHD);
  int j = eia[l], i = eia[L + l];
  float nrm = dis[j] * dis[i];
  atomicAdd(&out[(size_t)i * D_EHD + d], nrm * x[(size_t)j * D_EHD + d]);
}
// highway: dst = leaky( gate*relu(gcn) + (1-gate)*x1 )   (dst may alias x1)
__global__ void dsea_highway(const float* __restrict__ x1, const float* __restrict__ gcn,
                             const float* __restrict__ gate, float* __restrict__ dst, long long n){
  long long i = (long long)blockIdx.x * blockDim.x + threadIdx.x;
  if (i >= n) return;
  float x2 = fmaxf(gcn[i], 0.f);
  float g  = gate[i];
  dst[i] = dsea_leaky(g * x2 + (1.f - g) * x1[i]);
}
// out[m*ostride+ooff] = act( dot(A[m,:],v) + bias )
__global__ void dsea_rowdot(const float* __restrict__ A, int lda, const float* __restrict__ v,
                            int K, int M, float* __restrict__ out, int ostride, int ooff,
                            const float* __restrict__ biasp, int act){
  int m = blockIdx.x * blockDim.x + threadIdx.x;
  if (m >= M) return;
  const float* a = A + (size_t)m * lda;
  float s = biasp ? biasp[0] : 0.f;
  for (int k = 0; k < K; ++k) s += a[k] * v[k];
  out[(size_t)m * ostride + ooff] = dsea_act(s, act);
}
// gat_e edge scores: e1=leaky(sA[h]+sB[t]); e2=leaky(sC[h]+sD[t])
__global__ void dsea_edge2(const int* __restrict__ ei, int E,
                           const float* sA, const float* sB, const float* sC, const float* sD,
                           float* e1, float* e2){
  int l = blockIdx.x * blockDim.x + threadIdx.x;
  if (l >= E) return;
  int h = ei[l], t = ei[E + l];
  e1[l] = dsea_leaky(sA[h] + sB[t]);
  e2[l] = dsea_leaky(sC[h] + sD[t]);
}
// segment softmax pieces
__global__ void dsea_segmax(const float* e, const int* seg, float* m, int L){
  int l = blockIdx.x * blockDim.x + threadIdx.x;
  if (l < L) dsea_atomic_max_f(&m[seg[l]], e[l]);
}
__global__ void dsea_segexp(float* e, const int* seg, const float* m, float* s, int L){
  int l = blockIdx.x * blockDim.x + threadIdx.x;
  if (l >= L) return;
  float ex = expf(e[l] - m[seg[l]]);
  e[l] = ex;
  atomicAdd(&s[seg[l]], ex);
}
__global__ void dsea_segnorm(float* e, const int* seg, const float* s, int L){
  int l = blockIdx.x * blockDim.x + threadIdx.x;
  if (l < L) e[l] = e[l] / (s[seg[l]] + 1e-16f);
}
// gat_e scatter: xrh[rel,:] += a1*xh[h,:];  xrt[rel,:] += a2*xt[t,:]
__global__ void dsea_scatter_ge(const int* __restrict__ ei, const int* __restrict__ rel, int E,
                                const float* a1, const float* a2,
                                const float* __restrict__ xh, const float* __restrict__ xt,
                                float* __restrict__ xrh, float* __restrict__ xrt){
  long long tid = (long long)blockIdx.x * blockDim.x + threadIdx.x;
  if (tid >= (long long)E * D_RHD) return;
  int l = (int)(tid / D_RHD), d = (int)(tid - (long long)l * D_RHD);
  int r = rel[l], h = ei[l], t = ei[E + l];
  atomicAdd(&xrh[(size_t)r * D_RHD + d], a1[l] * xh[(size_t)h * D_RHD + d]);
  atomicAdd(&xrt[(size_t)r * D_RHD + d], a2[l] * xt[(size_t)t * D_RHD + d]);
}
__global__ void dsea_addinto(float* a, const float* b, long long n){
  long long i = (long long)blockIdx.x * blockDim.x + threadIdx.x;
  if (i < n) a[i] += b[i];
}
// gat_r edge scores: e = leaky(se[h] + sr[rel])
__global__ void dsea_edge_gr(const int* ei, const int* rel, int E,
                             const float* se, const float* sr, float* e){
  int l = blockIdx.x * blockDim.x + threadIdx.x;
  if (l >= E) return;
  e[l] = dsea_leaky(se[ei[l]] + sr[rel[l]]);
}
// gat_r scatter: out[rel,:] += a*xe[t,:]
__global__ void dsea_scatter_gr(const int* __restrict__ ei, const int* __restrict__ rel, int E,
                                const float* a, const float* __restrict__ xe,
                                float* __restrict__ out){
  long long tid = (long long)blockIdx.x * blockDim.x + threadIdx.x;
  if (tid >= (long long)E * D_EHD) return;
  int l = (int)(tid / D_EHD), d = (int)(tid - (long long)l * D_EHD);
  int r = rel[l], t = ei[E + l];
  atomicAdd(&out[(size_t)r * D_EHD + d], a[l] * xe[(size_t)t * D_EHD + d]);
}
__global__ void dsea_copycols(float* __restrict__ dst, int ldd, int c0,
                              const float* __restrict__ src, int lds, int nc, int M){
  long long tid = (long long)blockIdx.x * blockDim.x + threadIdx.x;
  if (tid >= (long long)M * nc) return;
  int m = (int)(tid / nc), d = (int)(tid - (long long)m * nc);
  dst[(size_t)m * ldd + c0 + d] = src[(size_t)m * lds + d];
}
// gat(all-edge): e = leaky(si[i] + sj[j])
__global__ void dsea_edge_gat(const int* eia, int L, const float* si, const float* sj, float* e){
  int l = blockIdx.x * blockDim.x + threadIdx.x;
  if (l >= L) return;
  e[l] = dsea_leaky(si[eia[L + l]] + sj[eia[l]]);
}
// gat scatter (only first 300 cols needed): out[i,:300] += a*x[j,:300] with x ld=600
__global__ void dsea_scatter_gat(const int* __restrict__ eia, int L, const float* a,
                                 const float* __restrict__ x, float* __restrict__ out){
  long long tid = (long long)blockIdx.x * blockDim.x + threadIdx.x;
  if (tid >= (long long)L * D_EHD) return;
  int l = (int)(tid / D_EHD), d = (int)(tid - (long long)l * D_EHD);
  int j = eia[l], i = eia[L + l];
  atomicAdd(&out[(size_t)i * D_EHD + d], a[l] * x[(size_t)j * 600 + d]);
}
// x_final = 9 * [ xcat(600) | leaky(out300) ]
__global__ void dsea_write_x(const float* __restrict__ xcat, const float* __restrict__ out300,
                             float* __restrict__ X){
  long long tid = (long long)blockIdx.x * blockDim.x + threadIdx.x;
  if (tid >= (long long)D_N * 900) return;
  int n = (int)(tid / 900), d = (int)(tid - (long long)n * 900);
  float v = (d < 600) ? xcat[(size_t)n * 600 + d]
                      : dsea_leaky(out300[(size_t)n * D_EHD + (d - 600)]);
  X[tid] = 9.f * v;
}
// LSTM cell (gate order i,f,g,o); writes h both to state and to output slot column
__global__ void dsea_lstm_cell(const float* __restrict__ g, float* __restrict__ c,
                               float* __restrict__ h, float* __restrict__ outslot,
                               int out_ld, int out_col, int M, int H, int first){
  long long tid = (long long)blockIdx.x * blockDim.x + threadIdx.x;
  if (tid >= (long long)M * H) return;
  int n = (int)(tid / H), d = (int)(tid - (long long)n * H);
  const float* gr = g + (size_t)n * 4 * H;
  float ig = dsea_sig(gr[d]);
  float fg = dsea_sig(gr[H + d]);
  float gg = tanhf(gr[2 * H + d]);
  float og = dsea_sig(gr[3 * H + d]);
  float cold = first ? 0.f : c[tid];
  float cnew = fg * cold + ig * gg;
  float hnew = og * tanhf(cnew);
  c[tid] = cnew;
  h[tid] = hnew;
  outslot[(size_t)n * out_ld + out_col + d] = hnew;
}
__global__ void dsea_gather(const float* __restrict__ src, const int* __restrict__ db, int col,
                            float* __restrict__ dst, int B, int Dd){
  long long tid = (long long)blockIdx.x * blockDim.x + threadIdx.x;
  if (tid >= (long long)B * Dd) return;
  int b = (int)(tid / Dd), d = (int)(tid - (long long)b * Dd);
  dst[tid] = src[(size_t)db[b * 2 + col] * Dd + d];
}
__global__ void dsea_row_softmax(float* __restrict__ x, int Dd){
  int row = blockIdx.x;
  float* p = x + (size_t)row * Dd;
  __shared__ float red[256];
  int tid = threadIdx.x;
  float m = -3.402823466e38f;
  for (int d = tid; d < Dd; d += 256) m = fmaxf(m, p[d]);
  red[tid] = m; __syncthreads();
  for (int s = 128; s > 0; s >>= 1) { if (tid < s) red[tid] = fmaxf(red[tid], red[tid + s]); __syncthreads(); }
  m = red[0]; __syncthreads();
  float sum = 0.f;
  for (int d = tid; d < Dd; d += 256) sum += expf(p[d] - m);
  red[tid] = sum; __syncthreads();
  for (int s = 128; s > 0; s >>= 1) { if (tid < s) red[tid] += red[tid + s]; __syncthreads(); }
  sum = red[0];
  float inv = 1.f / sum;
  for (int d = tid; d < Dd; d += 256) p[d] = expf(p[d] - m) * inv;
}
__global__ void dsea_softmax3(const float* sc, float* kg, int B){
  int b = blockIdx.x * blockDim.x + threadIdx.x;
  if (b >= B) return;
  float a0 = sc[b*3], a1 = sc[b*3+1], a2 = sc[b*3+2];
  float m = fmaxf(a0, fmaxf(a1, a2));
  float e0 = expf(a0-m), e1 = expf(a1-m), e2 = expf(a2-m);
  float s = e0 + e1 + e2;
  kg[b*3] = e0/s; kg[b*3+1] = e1/s; kg[b*3+2] = e2/s;
}
__global__ void dsea_winner(const int* db, int col, int* win, int B){
  int b = blockIdx.x * blockDim.x + threadIdx.x;
  if (b < B) atomicMax(&win[db[b*2+col]], b);
}
__global__ void dsea_apply_kg(float* __restrict__ nm, float* __restrict__ oh,
                              const int* db, int col, const int* win,
                              const float* kg, int B){
  long long tid = (long long)blockIdx.x * blockDim.x + threadIdx.x;
  if (tid >= (long long)B * 900) return;
  int b = (int)(tid / 900), d = (int)(tid - (long long)b * 900);
  int n = db[b*2+col];
  if (win[n] == b) {
    nm[(size_t)n*900 + d] *= kg[b*3+1];
    oh[(size_t)n*900 + d] *= kg[b*3+2];
  }
}

// ---------------- host ----------------
static inline int dsea_cdiv(long long a, long long b){ return (int)((a + b - 1) / b); }

extern "C" void kernel_launch(void* const* d_in, const int* in_sizes, int n_in,
                              void* d_out, int out_size, void* d_ws, size_t ws_size,
                              hipStream_t stream)
{
  (void)in_sizes; (void)n_in; (void)out_size; (void)ws_size;
  const float* x_e1   = (const float*)d_in[0];
  const int*   ei1    = (const int*)  d_in[1];
  const int*   rel1   = (const int*)  d_in[2];
  const int*   eia1   = (const int*)  d_in[3];
  const float* xname1 = (const float*)d_in[5];
  const float* oneh1  = (const float*)d_in[6];
  const float* x_e2   = (const float*)d_in[7];
  const int*   ei2    = (const int*)  d_in[8];
  const int*   rel2   = (const int*)  d_in[9];
  const int*   eia2   = (const int*)  d_in[10];
  const float* xname2 = (const float*)d_in[12];
  const float* oneh2  = (const float*)d_in[13];
  const int*   dbatch = (const int*)  d_in[14];
  const float* hw1_w  = (const float*)d_in[15];
  const float* hw2_w  = (const float*)d_in[16];
  const float* ge_wh  = (const float*)d_in[17];
  const float* ge_wt  = (const float*)d_in[18];
  const float* ge_ah1 = (const float*)d_in[19];
  const float* ge_ah2 = (const float*)d_in[20];
  const float* ge_at1 = (const float*)d_in[21];
  const float* ge_at2 = (const float*)d_in[22];
  const float* gr_ah  = (const float*)d_in[23];
  const float* gr_ar  = (const float*)d_in[24];
  const float* g_ai   = (const float*)d_in[25];
  const float* g_aj   = (const float*)d_in[26];
  const float* l1_wih_f = (const float*)d_in[27];
  const float* l1_whh_f = (const float*)d_in[28];
  const float* l1_bih_f = (const float*)d_in[29];
  const float* l1_bhh_f = (const float*)d_in[30];
  const float* l1_wih_b = (const float*)d_in[31];
  const float* l1_whh_b = (const float*)d_in[32];
  const float* l1_bih_b = (const float*)d_in[33];
  const float* l1_bhh_b = (const float*)d_in[34];
  const float* l2_wih   = (const float*)d_in[35];
  const float* l2_whh   = (const float*)d_in[36];
  const float* l2_bih   = (const float*)d_in[37];
  const float* l2_bhh   = (const float*)d_in[38];
  const float* mk_w1    = (const float*)d_in[39];
  const float* mk_b1    = (const float*)d_in[40];
  const float* mk_w2    = (const float*)d_in[41];
  const float* mk_b2    = (const float*)d_in[42];

  float* outf = (float*)d_out;
  const size_t NS = (size_t)D_N * 900;
  float *X1 = outf,          *NM1 = outf + NS,   *OH1 = outf + 2*NS;
  float *X2 = outf + 3*NS,   *NM2 = outf + 4*NS, *OH2 = outf + 5*NS;

  // ------- workspace layout (all region sizes even -> 8B alignment holds) -------
  float* wsf = (float*)d_ws;
  size_t off = 0;
  auto alloc = [&](size_t n){ float* p = wsf + off; off += n; return p; };
  float* xe    = alloc((size_t)D_N * 300);
  float* gcnb  = alloc((size_t)D_N * 300);
  float* gateb = alloc((size_t)D_N * 300);
  float* xh    = alloc((size_t)D_N * 100);
  float* xt    = alloc((size_t)D_N * 100);
  float* sA = alloc(D_N); float* sB = alloc(D_N); float* sC = alloc(D_N); float* sD = alloc(D_N);
  float* e1 = alloc(D_E); float* e2 = alloc(D_E);
  float* mR = alloc(D_R); float* sR = alloc(D_R);
  float* xrh = alloc((size_t)D_R * 100); float* xrt = alloc((size_t)D_R * 100);
  float* grN = alloc(D_N); float* grR = alloc(D_R);
  float* eall = alloc(D_EA);
  float* mN = alloc(D_N); float* sN = alloc(D_N);
  float* xcat = alloc((size_t)D_N * 600);
  float* hA = alloc((size_t)D_N * 300);
  float* cA = alloc((size_t)D_N * 300);
  float* bs1f = alloc(600); float* bs1b = alloc(600); float* bs2 = alloc(1200);
  float* scores = alloc((size_t)D_B * D_N);
  float* Xq  = alloc((size_t)D_B * 900);
  float* Re  = alloc((size_t)D_B * 900);
  float* h1mk = alloc((size_t)D_B * 450);
  float* scb = alloc((size_t)D_B * 3 + 2);
  float* kg1 = alloc((size_t)D_B * 3 + 2);
  float* kg2 = alloc((size_t)D_B * 3 + 2);
  int* win1 = (int*)alloc(D_N);
  int* win2 = (int*)alloc(D_N);
  // LSTM gates buffer (N x 1200 max) aliases dead graph scratch (xe..xcat span)
  float* gates = xe;

  auto fillf = [&](float* p, float v, long long n){
    dsea_fill<<<dsea_cdiv(n,256),256,0,stream>>>(p, v, n);
  };
  auto gemmNT = [&](int act, const float* A, int lda, const float* Bm, int ldb,
                    float* C, int ldc, const float* bias, const float* add, int ldadd,
                    float cadd, int M, int Nc, int K){
    int groups = ((M+15)/16) * (((Nc+15)/16 + 3) / 4);
    int blocks = (groups + 7) / 8;
    switch (act) {
      case 0: dsea_gemm<0,0><<<blocks,256,0,stream>>>(A,lda,Bm,ldb,C,ldc,bias,add,ldadd,cadd,M,Nc,K); break;
      case 1: dsea_gemm<0,1><<<blocks,256,0,stream>>>(A,lda,Bm,ldb,C,ldc,bias,add,ldadd,cadd,M,Nc,K); break;
      default:dsea_gemm<0,2><<<blocks,256,0,stream>>>(A,lda,Bm,ldb,C,ldc,bias,add,ldadd,cadd,M,Nc,K); break;
    }
  };
  auto gemmNN = [&](const float* A, int lda, const float* Bm, int ldb,
                    float* C, int ldc, const float* bias, const float* add, int ldadd,
                    float cadd, int M, int Nc, int K){
    int groups = ((M+15)/16) * (((Nc+15)/16 + 3) / 4);
    int blocks = (groups + 7) / 8;
    dsea_gemm<1,0><<<blocks,256,0,stream>>>(A,lda,Bm,ldb,C,ldc,bias,add,ldadd,cadd,M,Nc,K);
  };
  auto segsm = [&](float* e, const int* seg, int L, float* m, float* s, int num){
    fillf(m, -3.402823466e38f, num);
    fillf(s, 0.f, num);
    dsea_segmax <<<dsea_cdiv(L,256),256,0,stream>>>(e, seg, m, L);
    dsea_segexp <<<dsea_cdiv(L,256),256,0,stream>>>(e, seg, m, s, L);
    dsea_segnorm<<<dsea_cdiv(L,256),256,0,stream>>>(e, seg, s, L);
  };
  auto rowdot = [&](const float* A, int lda, const float* v, int K, int M,
                    float* out, int ostride, int ooff, const float* biasp, int act){
    dsea_rowdot<<<dsea_cdiv(M,256),256,0,stream>>>(A,lda,v,K,M,out,ostride,ooff,biasp,act);
  };
  auto lstm = [&](const float* seqbase, const float* wih, const float* whh, const float* bsum,
                  int H, float* outslot, int outoff, bool bwd){
    int G = 4 * H;
    for (int st = 0; st < 3; ++st) {
      int t = bwd ? (2 - st) : st;
      gemmNT(0, seqbase + (size_t)t * D_EHD, 900, wih, 300, gates, G,
             bsum, nullptr, 0, 0.f, D_N, G, 300);
      if (st > 0)
        gemmNT(0, hA, H, whh, H, gates, G, nullptr, gates, G, 0.f, D_N, G, H);
      dsea_lstm_cell<<<dsea_cdiv((long long)D_N*H,256),256,0,stream>>>(
          gates, cA, hA, outslot, 900, t * 300 + outoff, D_N, H, st == 0 ? 1 : 0);
    }
  };

  auto process_side = [&](const float* xin, const int* ei, const int* rel, const int* eia,
                          const float* xname, const float* onehot,
                          float* Xs, float* NMs, float* OHs){
    // degree / normalization (shared by both GCN layers)
    fillf(sA, 0.f, D_N);
    dsea_count<<<dsea_cdiv(D_EA,256),256,0,stream>>>(eia + D_EA, sA, D_EA);
    dsea_dis<<<dsea_cdiv(D_N,256),256,0,stream>>>(sA, sB, D_N);
    // layer 1
    fillf(gcnb, 0.f, (long long)D_N*300);
    dsea_gcn_scatter<<<dsea_cdiv((long long)D_EA*300,256),256,0,stream>>>(xin, eia, sB, gcnb, D_EA);
    gemmNT(2, xin, 300, hw1_w, 300, gateb, 300, nullptr, nullptr, 0, 1e-8f, D_N, 300, 300);
    dsea_highway<<<dsea_cdiv((long long)D_N*300,256),256,0,stream>>>(xin, gcnb, gateb, xe, (long long)D_N*300);
    // layer 2
    fillf(gcnb, 0.f, (long long)D_N*300);
    dsea_gcn_scatter<<<dsea_cdiv((long long)D_EA*300,256),256,0,stream>>>(xe, eia, sB, gcnb, D_EA);
    gemmNT(2, xe, 300, hw2_w, 300, gateb, 300, nullptr, nullptr, 0, 1e-8f, D_N, 300, 300);
    dsea_highway<<<dsea_cdiv((long long)D_N*300,256),256,0,stream>>>(xe, gcnb, gateb, xe, (long long)D_N*300);
    // gat_e
    gemmNT(0, xe, 300, ge_wh, 300, xh, 100, nullptr, nullptr, 0, 0.f, D_N, 100, 300);
    gemmNT(0, xe, 300, ge_wt, 300, xt, 100, nullptr, nullptr, 0, 0.f, D_N, 100, 300);
    rowdot(xh, 100, ge_ah1, 100, D_N, sA, 1, 0, nullptr, 0);
    rowdot(xt, 100, ge_ah2, 100, D_N, sB, 1, 0, nullptr, 0);
    rowdot(xh, 100, ge_at1, 100, D_N, sC, 1, 0, nullptr, 0);
    rowdot(xt, 100, ge_at2, 100, D_N, sD, 1, 0, nullptr, 0);
    dsea_edge2<<<dsea_cdiv(D_E,256),256,0,stream>>>(ei, D_E, sA, sB, sC, sD, e1, e2);
    segsm(e1, rel, D_E, mR, sR, D_R);
    segsm(e2, rel, D_E, mR, sR, D_R);
    fillf(xrh, 0.f, (long long)D_R*100);
    fillf(xrt, 0.f, (long long)D_R*100);
    dsea_scatter_ge<<<dsea_cdiv((long long)D_E*100,256),256,0,stream>>>(ei, rel, D_E, e1, e2, xh, xt, xrh, xrt);
    dsea_addinto<<<dsea_cdiv((long long)D_R*100,256),256,0,stream>>>(xrh, xrt, (long long)D_R*100); // x_r
    // gat_r
    rowdot(xe, 300, gr_ah, 300, D_N, grN, 1, 0, nullptr, 0);
    rowdot(xrh, 100, gr_ar, 100, D_R, grR, 1, 0, nullptr, 0);
    dsea_edge_gr<<<dsea_cdiv(D_E,256),256,0,stream>>>(ei, rel, D_E, grN, grR, e1);
    segsm(e1, rel, D_E, mR, sR, D_R);
    fillf(gcnb, 0.f, (long long)D_N*300); // out2 (N x 300, rows >= R stay zero)
    dsea_scatter_gr<<<dsea_cdiv((long long)D_E*300,256),256,0,stream>>>(ei, rel, D_E, e1, xe, gcnb);
    dsea_copycols<<<dsea_cdiv((long long)D_N*300,256),256,0,stream>>>(xcat, 600, 0,   xe,   300, 300, D_N);
    dsea_copycols<<<dsea_cdiv((long long)D_N*300,256),256,0,stream>>>(xcat, 600, 300, gcnb, 300, 300, D_N);
    // gat over all edges
    rowdot(xcat, 600, g_ai, 600, D_N, sA, 1, 0, nullptr, 0); // si (targets)
    rowdot(xcat, 600, g_aj, 600, D_N, sB, 1, 0, nullptr, 0); // sj (sources)
    dsea_edge_gat<<<dsea_cdiv(D_EA,256),256,0,stream>>>(eia, D_EA, sA, sB, eall);
    segsm(eall, eia + D_EA, D_EA, mN, sN, D_N);
    fillf(gateb, 0.f, (long long)D_N*300); // out300
    dsea_scatter_gat<<<dsea_cdiv((long long)D_EA*300,256),256,0,stream>>>(eia, D_EA, eall, xcat, gateb);
    dsea_write_x<<<dsea_cdiv((long long)D_N*900,256),256,0,stream>>>(xcat, gateb, Xs);
    // LSTMs (reuse dead graph scratch as gates buffer)
    lstm(xname, l1_wih_f, l1_whh_f, bs1f, 150, NMs, 0,   false);
    lstm(xname, l1_wih_b, l1_whh_b, bs1b, 150, NMs, 150, true);
    lstm(onehot, l2_wih,  l2_whh,  bs2, 300, OHs, 0,    false);
  };

  auto attend = [&](float* const* qs, float* const* ks2, int col, float* kg){
    for (int c = 0; c < 3; ++c) {
      dsea_gather<<<dsea_cdiv((long long)D_B*900,256),256,0,stream>>>(qs[c], dbatch, col, Xq, D_B, 900);
      gemmNT(0, Xq, 900, ks2[c], 900, scores, D_N, nullptr, nullptr, 0, 0.f, D_B, D_N, 900);
      dsea_row_softmax<<<D_B,256,0,stream>>>(scores, D_N);
      gemmNN(scores, D_N, ks2[c], 900, Re, 900, nullptr, Xq, 900, 0.f, D_B, 900, D_N);
      gemmNT(1, Re, 900, mk_w1, 900, h1mk, 450, mk_b1, nullptr, 0, 0.f, D_B, 450, 900);
      rowdot(h1mk, 450, mk_w2, 450, D_B, scb, 3, c, mk_b2, 3);
    }
    dsea_softmax3<<<dsea_cdiv(D_B,256),256,0,stream>>>(scb, kg, D_B);
  };

  // combined LSTM biases (bih + bhh)
  dsea_vecadd2<<<dsea_cdiv(600,256),256,0,stream>>>(l1_bih_f, l1_bhh_f, bs1f, 600);
  dsea_vecadd2<<<dsea_cdiv(600,256),256,0,stream>>>(l1_bih_b, l1_bhh_b, bs1b, 600);
  dsea_vecadd2<<<dsea_cdiv(1200,256),256,0,stream>>>(l2_bih, l2_bhh, bs2, 1200);

  process_side(x_e1, ei1, rel1, eia1, xname1, oneh1, X1, NM1, OH1);
  process_side(x_e2, ei2, rel2, eia2, xname2, oneh2, X2, NM2, OH2);

  float* q1[3] = {X1, NM1, OH1};
  float* q2[3] = {X2, NM2, OH2};
  attend(q1, q2, 0, kg1); // queries side1 (db0) vs keys side2
  attend(q2, q1, 1, kg2); // queries side2 (db1) vs keys side1

  dsea_fill_i<<<dsea_cdiv(D_N,256),256,0,stream>>>(win1, -1, D_N);
  dsea_fill_i<<<dsea_cdiv(D_N,256),256,0,stream>>>(win2, -1, D_N);
  dsea_winner<<<dsea_cdiv(D_B,256),256,0,stream>>>(dbatch, 0, win1, D_B);
  dsea_winner<<<dsea_cdiv(D_B,256),256,0,stream>>>(dbatch, 1, win2, D_B);
  dsea_apply_kg<<<dsea_cdiv((long long)D_B*900,256),256,0,stream>>>(NM1, OH1, dbatch, 0, win1, kg1, D_B);
  dsea_apply_kg<<<dsea_cdiv((long long)D_B*900,256),256,0,stream>>>(NM2, OH2, dbatch, 1, win2, kg2, D_B);

  dsea_row_softmax<<<D_N,256,0,stream>>>(NM1, 900);
  dsea_row_softmax<<<D_N,256,0,stream>>>(OH1, 900);
  dsea_row_softmax<<<D_N,256,0,stream>>>(NM2, 900);
  dsea_row_softmax<<<D_N,256,0,stream>>>(OH2, 900);
}